// OneHotWeightedAverage_74234214744157
// MI455X (gfx1250) — compile-verified
//
#include <hip/hip_runtime.h>

typedef float v2f __attribute__((ext_vector_type(2)));
typedef float v8f __attribute__((ext_vector_type(8)));
typedef int   v4i __attribute__((ext_vector_type(4)));

#define AS1 __attribute__((address_space(1)))
#define AS3 __attribute__((address_space(3)))

#define B_DIM 128
#define L_DIM 200
#define D_DIM 256
#define V_SIZE 50000
#define ROWS (B_DIM * L_DIM)          // 25600 flattened (b,l) rows
#define TILES (ROWS / 16)             // 1600 tiles of 16 rows
#define TILES_PER_BLOCK 2
#define LDS_STRIDE 260                // 256 + 4 pad: stride%64==4 -> conflict-free A reads

// ---------------------------------------------------------------------------
// Kernel 1: zero the [B, V] table with wide stores (pure bandwidth).
// ---------------------------------------------------------------------------
__global__ __launch_bounds__(256) void zero_kernel(float4* __restrict__ p, int n4) {
    int i = blockIdx.x * blockDim.x + threadIdx.x;
    if (i < n4) p[i] = make_float4(0.f, 0.f, 0.f, 0.f);
}

// ---------------------------------------------------------------------------
// Kernel 2: scores[r] = dot(w_es[r, :], score_v) via V_WMMA_F32_16X16X4_F32.
// One wave per 16-row tile; tile is a contiguous 16KB chunk of w_es, staged
// into LDS with GLOBAL_LOAD_ASYNC_TO_LDS_B128 (ASYNCcnt path, no VGPR hop).
// B-matrix columns all carry score_v -> every D column equals the score.
// ---------------------------------------------------------------------------
__global__ __launch_bounds__(64) void score_wmma_kernel(const float* __restrict__ w_es,
                                                        const float* __restrict__ score_v,
                                                        float* __restrict__ scores) {
    __shared__ float s_sv[D_DIM];
    __shared__ float s_tile[TILES_PER_BLOCK][16 * LDS_STRIDE];

    const int tid  = threadIdx.x;
    const int lane = tid & 31;
    const int wave = tid >> 5;

    // stage score_v (1 KB, shared by both waves)
    #pragma unroll
    for (int i = tid; i < D_DIM; i += 64) s_sv[i] = score_v[i];

    const int tile = blockIdx.x * TILES_PER_BLOCK + wave;

    // stage 16x256 f32 tile (contiguous 4096 floats) with padded row stride,
    // directly into LDS via the CDNA5 async-copy path.
    const float* __restrict__ gsrc = w_es + (size_t)tile * 16 * D_DIM;
    float* __restrict__ dst = s_tile[wave];

#if __has_builtin(__builtin_amdgcn_global_load_async_to_lds_b128)
    #pragma unroll
    for (int j = 0; j < 32; ++j) {
        int idx4 = lane + 32 * j;            // float4 index within tile
        int row  = idx4 >> 6;                // 64 float4 per 256-float row
        int col  = (idx4 & 63) * 4;
        __builtin_amdgcn_global_load_async_to_lds_b128(
            (AS1 v4i*)(gsrc + idx4 * 4),
            (AS3 v4i*)(dst + row * LDS_STRIDE + col),
            0, 0);
    }
  #if __has_builtin(__builtin_amdgcn_s_wait_asynccnt)
    __builtin_amdgcn_s_wait_asynccnt(0);
  #else
    asm volatile("s_wait_asynccnt 0x0" ::: "memory");
  #endif
#else
    // Inline-asm fallback: generic LDS address low 32 bits == LDS byte offset
    // (ISA 10.2 aperture rule), so no address-space cast is needed.
    #pragma unroll
    for (int j = 0; j < 32; ++j) {
        int idx4 = lane + 32 * j;
        int row  = idx4 >> 6;
        int col  = (idx4 & 63) * 4;
        unsigned           lds_addr = (unsigned)(__SIZE_TYPE__)(const void*)(dst + row * LDS_STRIDE + col);
        unsigned long long gaddr    = (unsigned long long)(gsrc + idx4 * 4);
        asm volatile("global_load_async_to_lds_b128 %0, %1, off"
                     :: "v"(lds_addr), "v"(gaddr) : "memory");
    }
    asm volatile("s_wait_asynccnt 0x0" ::: "memory");
#endif
    __syncthreads();

    // A-fragment layout (32-bit A 16x4): lane m, half h -> K = {k+2h, k+2h+1}
    const int hi = lane >> 4;               // half-wave
    const int m  = lane & 15;               // A row
    const float* __restrict__ arow = dst + m * LDS_STRIDE + 2 * hi;
    const float* __restrict__ brow = s_sv + 2 * hi;

    v8f acc = {};
    #pragma unroll 4
    for (int k = 0; k < D_DIM; k += 4) {
        v2f a; a.x = arow[k];  a.y = arow[k + 1];
        v2f b; b.x = brow[k];  b.y = brow[k + 1];   // broadcast across N columns
        acc = __builtin_amdgcn_wmma_f32_16x16x4_f32(false, a, false, b,
                                                    (short)0, acc, false, false);
    }

    // C/D layout: VGPR v -> M = v (lanes 0-15) / v+8 (lanes 16-31); lane = N.
    // All N columns identical; lane with N==0 in each half writes 8 scores.
    if ((lane & 15) == 0) {
        float* __restrict__ out = scores + tile * 16 + 8 * hi;
        #pragma unroll
        for (int v = 0; v < 8; ++v) out[v] = acc[v];
    }
}

// ---------------------------------------------------------------------------
// Kernel 3: per-batch softmax over L=200 + scatter into w_a, write weights.
// ---------------------------------------------------------------------------
__global__ __launch_bounds__(256) void softmax_scatter_kernel(const float* __restrict__ scores,
                                                              const int* __restrict__ x,
                                                              float* __restrict__ w_a,
                                                              float* __restrict__ weights_out) {
    __shared__ float red[256];
    const int b = blockIdx.x;
    const int t = threadIdx.x;

    const float s = (t < L_DIM) ? scores[b * L_DIM + t] : -__builtin_inff();

    // max reduction
    red[t] = s;
    __syncthreads();
    #pragma unroll
    for (int off = 128; off > 0; off >>= 1) {
        if (t < off) red[t] = fmaxf(red[t], red[t + off]);
        __syncthreads();
    }
    const float mx = red[0];
    __syncthreads();

    const float e = (t < L_DIM) ? __expf(s - mx) : 0.f;

    // sum reduction
    red[t] = e;
    __syncthreads();
    #pragma unroll
    for (int off = 128; off > 0; off >>= 1) {
        if (t < off) red[t] += red[t + off];
        __syncthreads();
    }
    const float inv_sum = 1.f / red[0];

    if (t < L_DIM) {
        const float w = e * inv_sum;
        weights_out[b * L_DIM + t] = w;
        const int idx = x[b * L_DIM + t];
        w_a[(size_t)b * V_SIZE + idx] = w;
    }
}

// ---------------------------------------------------------------------------
extern "C" void kernel_launch(void* const* d_in, const int* in_sizes, int n_in,
                              void* d_out, int out_size, void* d_ws, size_t ws_size,
                              hipStream_t stream) {
    const float* w_es    = (const float*)d_in[0];   // [B, L, D] f32
    const float* score_v = (const float*)d_in[1];   // [D] f32
    const int*   x       = (const int*)d_in[2];     // [B, L] int

    float* out     = (float*)d_out;
    float* w_a     = out;                               // [B, V]
    float* weights = out + (size_t)B_DIM * V_SIZE;      // [B, L]
    float* scores  = (float*)d_ws;                      // [B*L] scratch

    const int n4 = (B_DIM * V_SIZE) / 4;                // 1,600,000 float4s
    zero_kernel<<<(n4 + 255) / 256, 256, 0, stream>>>((float4*)w_a, n4);

    score_wmma_kernel<<<TILES / TILES_PER_BLOCK, 64, 0, stream>>>(w_es, score_v, scores);

    softmax_scatter_kernel<<<B_DIM, 256, 0, stream>>>(scores, x, w_a, weights);
}